// HierarchicalSoftmaxLoss_18932215841020
// MI455X (gfx1250) — compile-verified
//
#include <hip/hip_runtime.h>
#include <hip/hip_bf16.h>
#include <math.h>

typedef float v2f __attribute__((ext_vector_type(2)));
typedef float v8f __attribute__((ext_vector_type(8)));

// softplus(x) = log(1 + e^x), numerically stable
__device__ __forceinline__ float softplus_f(float x) {
    return fmaxf(x, 0.0f) + log1pf(__expf(-fabsf(x)));
}

// ---------------------------------------------------------------------------
// Kernel 1: per-sample tree walk + deterministic per-block partial sum.
// One thread per sample (grid-stride for generality). All 16 tree indices are
// independent (popcount of the top-t bits), so we issue all 16 gathers before
// the first use -> 16 outstanding loads per thread hide HBM latency.
// ---------------------------------------------------------------------------
template <int L>
__global__ __launch_bounds__(256) void hsm_partial_fixed(
    const float* __restrict__ scores, const int* __restrict__ cls,
    float* __restrict__ partials, int B, int V) {
    float acc = 0.0f;
    const int stride = gridDim.x * blockDim.x;
    for (int b = blockIdx.x * blockDim.x + threadIdx.x; b < B; b += stride) {
        const unsigned ci = (unsigned)cls[b];
        const float* row = scores + (size_t)b * (size_t)V;
        float x[L];
#pragma unroll
        for (int t = 0; t < L; ++t) {
            const int idx = (1 << t) - 1 + (int)__popc(ci >> (L - t));
            const float sc = __builtin_nontemporal_load(row + idx);  // TH=NT: lines used once
            x[t] = ((ci >> (L - 1 - t)) & 1u) ? sc : -sc;
        }
#pragma unroll
        for (int t = 0; t < L; ++t) acc += softplus_f(x[t]);
    }
    // wave32 butterfly reduction
#pragma unroll
    for (int m = 16; m > 0; m >>= 1) acc += __shfl_xor(acc, m, 32);
    __shared__ float wsum[8];
    const int lane = threadIdx.x & 31;
    const int wid = threadIdx.x >> 5;
    if (lane == 0) wsum[wid] = acc;
    __syncthreads();
    if (threadIdx.x == 0) {
        float s = 0.0f;
#pragma unroll
        for (int w = 0; w < 8; ++w) s += wsum[w];  // fixed order -> deterministic
        partials[blockIdx.x] = s;
    }
}

// Fallback for dict sizes whose code length != 16 (not hit for V=50257).
__global__ __launch_bounds__(256) void hsm_partial_dyn(
    const float* __restrict__ scores, const int* __restrict__ cls,
    float* __restrict__ partials, int B, int V, int L) {
    float acc = 0.0f;
    const int stride = gridDim.x * blockDim.x;
    for (int b = blockIdx.x * blockDim.x + threadIdx.x; b < B; b += stride) {
        const unsigned ci = (unsigned)cls[b];
        const float* row = scores + (size_t)b * (size_t)V;
        for (int t = 0; t < L; ++t) {
            const int idx = (1 << t) - 1 + (int)__popc(ci >> (L - t));
            const float sc = __builtin_nontemporal_load(row + idx);
            const float x = ((ci >> (L - 1 - t)) & 1u) ? sc : -sc;
            acc += softplus_f(x);
        }
    }
#pragma unroll
    for (int m = 16; m > 0; m >>= 1) acc += __shfl_xor(acc, m, 32);
    __shared__ float wsum[8];
    const int lane = threadIdx.x & 31;
    const int wid = threadIdx.x >> 5;
    if (lane == 0) wsum[wid] = acc;
    __syncthreads();
    if (threadIdx.x == 0) {
        float s = 0.0f;
#pragma unroll
        for (int w = 0; w < 8; ++w) s += wsum[w];
        partials[blockIdx.x] = s;
    }
}

// ---------------------------------------------------------------------------
// Kernel 2: single wave32 reduces the 32 block partials deterministically.
// Reduction via V_WMMA_F32_16X16X4_F32 ones-trick: A = ones(16x4), B holds the
// 32 partials (K=0/K=1 rows across lane halves, other rows zero). Every row of
// D = column sums, so each lane gets partial[l%16] + partial[l%16 + 16]; a
// 4-step shfl_xor finishes the 16-wide sum. EXEC is all-ones (full wave, no
// divergence before the WMMA), as the ISA requires.
// ---------------------------------------------------------------------------
__global__ __launch_bounds__(32) void hsm_finalize(
    const float* __restrict__ partials, float* __restrict__ out,
    int nparts, float invB) {
    const int lane = threadIdx.x;  // exactly one wave32
    const float v = (lane < nparts) ? partials[lane] : 0.0f;
    float colsum;
#if __has_builtin(__builtin_amdgcn_wmma_f32_16x16x4_f32)
    v2f a;
    a.x = 1.0f;
    a.y = 1.0f;            // A = all-ones 16x4
    v2f bmat;
    bmat.x = v;
    bmat.y = 0.0f;         // B: one partial per lane, remaining K-rows zero
    v8f c = {};
    v8f d = __builtin_amdgcn_wmma_f32_16x16x4_f32(
        /*neg_a=*/false, a, /*neg_b=*/false, bmat,
        /*c_mod=*/(short)0, c, /*reuse_a=*/false, /*reuse_b=*/false);
    colsum = d[0];  // lane l: sum over K of B[:, l%16] = v[l%16] + v[l%16+16]
#else
    colsum = v + __shfl_xor(v, 16, 32);
#endif
    float t = colsum;
#pragma unroll
    for (int m = 8; m > 0; m >>= 1) t += __shfl_xor(t, m, 32);
    if (lane == 0) out[0] = t * invB;
}

extern "C" void kernel_launch(void* const* d_in, const int* in_sizes, int n_in,
                              void* d_out, int out_size, void* d_ws, size_t ws_size,
                              hipStream_t stream) {
    const float* scores = (const float*)d_in[0];
    const int* cls = (const int*)d_in[1];
    const int B = in_sizes[1];
    const int V = in_sizes[0] / B;
    int L = 0;
    while ((1u << L) < (unsigned)V) ++L;  // ceil(log2(V)); 16 for V=50257

    float* partials = (float*)d_ws;  // 32 floats of scratch
    constexpr int NB = 32, NT = 256; // 8192 threads = 1 thread per sample

    if (L == 16) {
        hsm_partial_fixed<16><<<NB, NT, 0, stream>>>(scores, cls, partials, B, V);
    } else {
        hsm_partial_dyn<<<NB, NT, 0, stream>>>(scores, cls, partials, B, V, L);
    }
    hsm_finalize<<<1, 32, 0, stream>>>(partials, (float*)d_out, NB, 1.0f / (float)B);
}